// Fast_Linear_44281112821805
// MI455X (gfx1250) — compile-verified
//
#include <hip/hip_runtime.h>

// ---------------------------------------------------------------------------
// relu(x @ w + b):  x[131072,4096] f32, w[4096,32] f32, b[32] f32
// Memory-bound (16 FLOP/byte, 2.15 GB of x => ~92us floor @ 23.3 TB/s).
// Exact f32 math via V_WMMA_F32_16X16X4_F32. Each wave owns a 32x32 output
// tile (two 16-row A stripes x both 16-col N tiles): per K-step
//   2x global_load_b64 (A) + 2x ds_load_b64 (B, pair-interleaved LDS) + 4 WMMA
// B operands land directly in aligned VGPR pairs (no repack movs).
// ---------------------------------------------------------------------------

typedef __attribute__((ext_vector_type(2))) float v2f;
typedef __attribute__((ext_vector_type(8))) float v8f;

#define BATCH   131072
#define IN_DIM  4096
#define UNITS   32
#define KCHUNK  256                    // K-rows of w staged per LDS refill
#define KPAIRS  (KCHUNK / 2)           // 128 pairs; LDS = 128*32*8B = 32 KB
#define WAVES_PER_BLOCK 8
#define ROWS_PER_WAVE   32
#define ROWS_PER_BLOCK  (WAVES_PER_BLOCK * ROWS_PER_WAVE)   // 256

__device__ __forceinline__ v8f wmma_f32(v2f a, v2f b, v8f c) {
    return __builtin_amdgcn_wmma_f32_16x16x4_f32(
        false, a, false, b, (short)0, c, false, false);
}

__global__ __launch_bounds__(256)
void fast_linear_wmma_f32(const float* __restrict__ x,
                          const float* __restrict__ w,
                          const float* __restrict__ b,
                          float* __restrict__ out) {
    // lds_wp[kp*32 + n] = { w[kc + 2*kp][n], w[kc + 2*kp + 1][n] }
    __shared__ v2f lds_wp[KPAIRS * UNITS];

    const int tid  = threadIdx.x;
    const int wave = tid >> 5;          // 0..7
    const int lane = tid & 31;          // wave32
    const int half = lane >> 4;         // K sub-pair select
    const int m    = lane & 15;         // M row / N col within tile

    const long row0 = (long)blockIdx.x * ROWS_PER_BLOCK + (long)wave * ROWS_PER_WAVE;

    // A-operand (16x4 f32): lane (m + 16*half) holds K = 2*half, 2*half+1.
    const float* xp0 = x + (row0 + m) * (long)IN_DIM + 2 * half;          // rows 0..15
    const float* xp1 = xp0 + 16 * (long)IN_DIM;                           // rows 16..31

    v8f acc00 = {};   // rows 0-15,  N 0-15
    v8f acc01 = {};   // rows 0-15,  N 16-31
    v8f acc10 = {};   // rows 16-31, N 0-15
    v8f acc11 = {};   // rows 16-31, N 16-31

    const int sn   = tid & 31;          // staging: n column (coalesced)
    const int skp0 = tid >> 5;          // staging: pair slot 0..7

    for (int kc = 0; kc < IN_DIM; kc += KCHUNK) {
        // --- stage w chunk into pair-interleaved LDS ---
        {
            const float* wsrc = w + (long)kc * UNITS;
            #pragma unroll
            for (int j = 0; j < KPAIRS / WAVES_PER_BLOCK; ++j) {   // 16 iters
                const int kp = skp0 + WAVES_PER_BLOCK * j;
                v2f p;
                p.x = wsrc[(2 * kp    ) * UNITS + sn];   // 128B-coalesced row
                p.y = wsrc[(2 * kp + 1) * UNITS + sn];
                lds_wp[kp * UNITS + sn] = p;             // conflict-free b64
            }
        }
        __syncthreads();

        // prefetch next x chunk for both stripes (global_prefetch_b8 path)
        if (kc + KCHUNK < IN_DIM) {
            __builtin_prefetch(xp0 + kc + KCHUNK, 0, 1);
            __builtin_prefetch(xp1 + kc + KCHUNK, 0, 1);
        }

        #pragma unroll 4
        for (int k = 0; k < KCHUNK; k += 4) {
            v2f a0 = *(const v2f*)(xp0 + kc + k);
            v2f a1 = *(const v2f*)(xp1 + kc + k);

            const int kp = (k >> 1) + half;       // this lane's K pair
            v2f b0 = lds_wp[kp * UNITS + m];      // N tile 0
            v2f b1 = lds_wp[kp * UNITS + m + 16]; // N tile 1

            acc00 = wmma_f32(a0, b0, acc00);
            acc01 = wmma_f32(a0, b1, acc01);
            acc10 = wmma_f32(a1, b0, acc10);
            acc11 = wmma_f32(a1, b1, acc11);
        }
        __syncthreads();
    }

    // --- epilogue: bias + relu + store ---
    // C/D layout: VGPR r, lanes 0-15 -> (M=r, N=lane), lanes 16-31 -> (M=8+r)
    const float bias0 = b[m];
    const float bias1 = b[m + 16];
    #pragma unroll
    for (int r = 0; r < 8; ++r) {
        const long orow0 = row0 + r + 8 * half;   // stripe 0
        const long orow1 = orow0 + 16;            // stripe 1
        float v00 = acc00[r] + bias0;  v00 = v00 > 0.0f ? v00 : 0.0f;
        float v01 = acc01[r] + bias1;  v01 = v01 > 0.0f ? v01 : 0.0f;
        float v10 = acc10[r] + bias0;  v10 = v10 > 0.0f ? v10 : 0.0f;
        float v11 = acc11[r] + bias1;  v11 = v11 > 0.0f ? v11 : 0.0f;
        out[orow0 * UNITS + m]      = v00;
        out[orow0 * UNITS + m + 16] = v01;
        out[orow1 * UNITS + m]      = v10;
        out[orow1 * UNITS + m + 16] = v11;
    }
}

extern "C" void kernel_launch(void* const* d_in, const int* in_sizes, int n_in,
                              void* d_out, int out_size, void* d_ws, size_t ws_size,
                              hipStream_t stream) {
    (void)in_sizes; (void)n_in; (void)out_size; (void)d_ws; (void)ws_size;
    const float* x = (const float*)d_in[0];
    const float* w = (const float*)d_in[1];
    const float* b = (const float*)d_in[2];
    float* out     = (float*)d_out;

    dim3 grid(BATCH / ROWS_PER_BLOCK);   // 512 blocks
    dim3 block(256);                     // 8 wave32 per block
    hipLaunchKernelGGL(fast_linear_wmma_f32, grid, block, 0, stream,
                       x, w, b, out);
}